// EKPConv_v2_35725537968374
// MI455X (gfx1250) — compile-verified
//
#include <hip/hip_runtime.h>

// ---------------------------------------------------------------------------
// EKPConv fused kernel for MI455X (gfx1250, wave32, WMMA).
//
// Roofline: ~15.7 GFLOP fp32 total, ~20 MB HBM traffic -> compute bound.
// The 9.8 GF terminal GEMM [10000 x 3840] @ [3840 x 128] and the 2.5 GF
// per-(query,lrf) aggregation GEMMs run on V_WMMA_F32_16X16X4_F32 (fp32 in,
// fp32 acc -> numerically matches the fp32 reference).
//
// One fused kernel: workgroup = 16 queries, wf[16][3840] lives in LDS
// (308 KB dynamic LDS, legal on CDNA5's 320 KB WGP LDS) so no global
// workspace is needed at all.
//
// Round-1 fixes: (a) LRF-feature half of the aggregation GEMM runs FIRST so
// feat[32] dies before any WMMA region (removes VGPR spills / vgpr_msb use);
// (b) LRF-MLP s-loop kept rolled to cut 1152 -> 288 static FMAs (I-cache).
// ---------------------------------------------------------------------------

typedef __attribute__((ext_vector_type(2))) float v2f;
typedef __attribute__((ext_vector_type(8))) float v8f;

#define NQ_TOT 10000
#define NS_TOT 10000
#define HN     32
#define KPTS   15
#define QB     16          // queries per workgroup (625 blocks exactly)
#define SWF    3844        // wf row stride (3840 + 4): conflict-free b64 A-frag loads
#define WA_STR 34          // w scratch stride (16 rows)
#define FB_STR 36          // f scratch stride (32 rows, float4-aligned writes)

#define OFF_MLP   (QB * SWF)          // 61504 : mlp_w [36*32]
#define OFF_MLPB  (OFF_MLP + 1152)    // 62656 : mlp_b [32]
#define OFF_KP    (OFF_MLPB + 32)     // 62688 : kernel_points [45] (pad 48)
#define OFF_SCR   (OFF_KP + 48)       // 62736 : per-wave scratch
#define SCR_WAVE  1792                // wA 16*34=544  +  fB 32*36=1152  (+pad)
#define SMEM_FLOATS (OFF_SCR + 8 * SCR_WAVE)   // 77072 floats = 308288 B

static __device__ __forceinline__ v8f wmma_f32(v2f a, v2f b, v8f c) {
  // D = A(16x4 f32) * B(4x16 f32) + C(16x16 f32)
  return __builtin_amdgcn_wmma_f32_16x16x4_f32(
      /*neg_a=*/false, a, /*neg_b=*/false, b,
      /*c_mod=*/(short)0, c, /*reuse_a=*/false, /*reuse_b=*/false);
}

// [16x32] @ [32x64] half (two 16-col tiles) with K=32 in 8 WMMA steps.
static __device__ __forceinline__ void mm_half(const v2f* afr, const float* fB,
                                               float* s_wf, int nl, int cbase,
                                               int rowhalf, int rcol, int kof) {
  v8f a0 = {0.f,0.f,0.f,0.f,0.f,0.f,0.f,0.f};
  v8f a1 = {0.f,0.f,0.f,0.f,0.f,0.f,0.f,0.f};
#pragma unroll
  for (int s8 = 0; s8 < 8; ++s8) {
    const float* bp = fB + (s8 * 4 + kof) * FB_STR;   // rows k0+kof, k0+kof+1
    v2f b0, b1;
    b0[0] = bp[rcol];                b0[1] = bp[FB_STR + rcol];
    b1[0] = bp[16 + rcol];           b1[1] = bp[FB_STR + 16 + rcol];
    a0 = wmma_f32(afr[s8], b0, a0);
    a1 = wmma_f32(afr[s8], b1, a1);
  }
  // D layout: VGPR i -> row i (lanes 0-15) / row 8+i (lanes 16-31), col = lane%16
#pragma unroll
  for (int i = 0; i < 8; ++i) {
    const int m = i + (rowhalf << 3);          // kernel-point index (row 15 = pad)
    if (m < KPTS) {
      s_wf[nl * SWF + m * 256 + cbase + rcol]      = a0[i];
      s_wf[nl * SWF + m * 256 + cbase + 16 + rcol] = a1[i];
    }
  }
}

__global__ __launch_bounds__(256, 1)
void ekpconv_fused(const float* __restrict__ q_pts,
                   const float* __restrict__ s_pts,
                   const int*   __restrict__ neighb,
                   const float* __restrict__ x,
                   const float* __restrict__ q_lrf,
                   const float* __restrict__ s_lrf,
                   const float* __restrict__ weights,       // [15][256][128]
                   const float* __restrict__ kernel_points, // [15][3]
                   const float* __restrict__ mlp_w,         // [36][32]
                   const float* __restrict__ mlp_b,         // [32]
                   float* __restrict__ out)                 // [10000][128]
{
  extern __shared__ float smem[];
  float* s_wf   = smem;                 // [16][SWF]  wf tiles
  float* s_mlp  = smem + OFF_MLP;
  float* s_mlpb = smem + OFF_MLPB;
  float* s_kp   = smem + OFF_KP;

  const int tid  = threadIdx.x;
  const int wave = tid >> 5;
  const int lane = tid & 31;
  float* wA = smem + OFF_SCR + wave * SCR_WAVE;  // [16][WA_STR] per-wave
  float* fB = wA + 16 * WA_STR;                  // [32][FB_STR] per-wave

  for (int i = tid; i < 1152; i += 256) s_mlp[i] = mlp_w[i];
  if (tid < 32) s_mlpb[tid] = mlp_b[tid];
  if (tid < KPTS * 3) s_kp[tid] = kernel_points[tid];
  __syncthreads();

  const int qblk    = blockIdx.x * QB;     // 625 * 16 == 10000 exactly
  const int rowhalf = lane >> 4;           // 0 | 1
  const int rcol    = lane & 15;
  const int kof     = rowhalf << 1;        // K sub-offset 0 | 2 (ISA frag layout)

  // ------------------------- Stage A: build wf in LDS ---------------------
#pragma unroll 1
  for (int pair = wave; pair < QB * 4; pair += 8) {
    const int nl = pair >> 2;              // local query
    const int q  = pair & 3;               // query-LRF index
    const int n  = qblk + nl;

    const int  ind0  = neighb[n * HN + lane];
    const bool valid = (unsigned)ind0 < (unsigned)NS_TOT;   // NS == shadow point
    const int  ind   = valid ? ind0 : 0;
    const float vm   = valid ? 1.0f : 0.0f;

    float R[9];                            // q_lrf[n][q] row-major [d][e]
#pragma unroll
    for (int j = 0; j < 9; ++j) R[j] = q_lrf[(n * 4 + q) * 9 + j];

    // -- geometry: rotate neighbor into LRF, 15 kernel-point weights -> wA --
    {
      const float qx = q_pts[n * 3 + 0], qy = q_pts[n * 3 + 1], qz = q_pts[n * 3 + 2];
      const float dx = s_pts[ind * 3 + 0] - qx;
      const float dy = s_pts[ind * 3 + 1] - qy;
      const float dz = s_pts[ind * 3 + 2] - qz;
      const float e0 = dx * R[0] + dy * R[3] + dz * R[6];
      const float e1 = dx * R[1] + dy * R[4] + dz * R[7];
      const float e2 = dx * R[2] + dy * R[5] + dz * R[8];
#pragma unroll
      for (int k = 0; k < 16; ++k) {       // row 15 = zero pad for WMMA M=16
        float w = 0.0f;
        if (k < KPTS) {
          const float t0 = e0 - s_kp[k * 3 + 0];
          const float t1 = e1 - s_kp[k * 3 + 1];
          const float t2 = e2 - s_kp[k * 3 + 2];
          const float d  = sqrtf(t0 * t0 + t1 * t1 + t2 * t2);
          w = fmaxf(0.0f, 1.0f - d * (1.0f / 1.2f)) * vm;
        }
        wA[k * WA_STR + lane] = w;         // A matrix: [k rows][h cols]
      }
    }

    // -- LRF feature MLP: feat[c] = b[c] + sum_j (R^T @ s_lrf)_j * W[j][c] --
    // s-loop kept rolled: 288 static FMAs instead of 1152 (I-cache).
    {
      float feat[32];
#pragma unroll
      for (int c = 0; c < 32; ++c) feat[c] = s_mlpb[c];
#pragma unroll 1
      for (int s = 0; s < 4; ++s) {
        float L[9];
#pragma unroll
        for (int j = 0; j < 9; ++j) L[j] = s_lrf[(ind * 4 + s) * 9 + j] * vm;
        const float* mwS = s_mlp + s * 288;
#pragma unroll
        for (int a = 0; a < 3; ++a)
#pragma unroll
          for (int c2 = 0; c2 < 3; ++c2) {
            const float m = R[a] * L[c2] + R[3 + a] * L[3 + c2] + R[6 + a] * L[6 + c2];
            const float* mw = mwS + (a * 3 + c2) * 32;
#pragma unroll
            for (int c = 0; c < 32; ++c) feat[c] = fmaf(m, mw[c], feat[c]);
          }
      }
      // stage masked features into fB NOW so feat[] dies before any WMMA
      float4* fp = reinterpret_cast<float4*>(fB + lane * FB_STR);
#pragma unroll
      for (int v = 0; v < 8; ++v) {
        float4 t;
        t.x = feat[v * 4 + 0] * vm; t.y = feat[v * 4 + 1] * vm;
        t.z = feat[v * 4 + 2] * vm; t.w = feat[v * 4 + 3] * vm;
        fp[v] = t;
      }
    }
    __builtin_amdgcn_wave_barrier();       // order LDS writes vs frag reads

    // A fragments for all 8 K-steps (K=32): lane holds row rcol, K=kof..kof+1
    v2f afr[8];
#pragma unroll
    for (int s8 = 0; s8 < 8; ++s8)
      afr[s8] = *reinterpret_cast<const v2f*>(wA + rcol * WA_STR + s8 * 4 + kof);

    // LRF half first (feat data currently in fB) -> wf cols q*64+32..q*64+63
    mm_half(afr, fB, s_wf, nl, q * 64 + 32, rowhalf, rcol, kof);

    // x half: gathered features (channel group q), masked, transient regs only
    {
      const float4* xp = reinterpret_cast<const float4*>(x + ind * 128 + q * 32);
      float4* fp = reinterpret_cast<float4*>(fB + lane * FB_STR);
#pragma unroll
      for (int v = 0; v < 8; ++v) {
        float4 t = xp[v];
        t.x *= vm; t.y *= vm; t.z *= vm; t.w *= vm;
        fp[v] = t;
      }
    }
    __builtin_amdgcn_wave_barrier();
    mm_half(afr, fB, s_wf, nl, q * 64, rowhalf, rcol, kof);
  }

  __syncthreads();

  // ------------- Stage B: out[16x128] = wf[16x3840] @ W[3840x128] ---------
  // Wave w owns output columns [16w, 16w+16). B streams from L2-resident
  // weights; each weight dword is consumed exactly once per block.
  {
    v8f acc = {0.f,0.f,0.f,0.f,0.f,0.f,0.f,0.f};
    const float* arow = s_wf + rcol * SWF;             // A row = local query
    const float* wcol = weights + wave * 16 + rcol;    // B col
#pragma unroll 8
    for (int k0 = 0; k0 < 3840; k0 += 4) {
      const v2f a = *reinterpret_cast<const v2f*>(arow + k0 + kof);
      v2f b;
      b[0] = wcol[(k0 + kof) * 128];
      b[1] = wcol[(k0 + kof + 1) * 128];
      acc = wmma_f32(a, b, acc);
    }
#pragma unroll
    for (int i = 0; i < 8; ++i) {
      const int m = i + (rowhalf << 3);                // local query row
      out[(qblk + m) * 128 + wave * 16 + rcol] = acc[i];
    }
  }
}

extern "C" void kernel_launch(void* const* d_in, const int* in_sizes, int n_in,
                              void* d_out, int out_size, void* d_ws, size_t ws_size,
                              hipStream_t stream) {
  (void)in_sizes; (void)n_in; (void)out_size; (void)d_ws; (void)ws_size;
  const float* q_pts   = (const float*)d_in[0];
  const float* s_pts   = (const float*)d_in[1];
  const int*   neighb  = (const int*)  d_in[2];
  const float* x       = (const float*)d_in[3];
  const float* q_lrf   = (const float*)d_in[4];
  const float* s_lrf   = (const float*)d_in[5];
  const float* weights = (const float*)d_in[6];
  const float* kp      = (const float*)d_in[7];
  const float* mlp_w   = (const float*)d_in[8];
  const float* mlp_b   = (const float*)d_in[9];
  float* out = (float*)d_out;

  const size_t shmem = (size_t)SMEM_FLOATS * sizeof(float);   // 308288 B
  // Opt in to >64KB dynamic LDS (CDNA5 WGP has 320KB). Deterministic; not a
  // stream op, so safe under graph capture.
  hipFuncSetAttribute(reinterpret_cast<const void*>(ekpconv_fused),
                      hipFuncAttributeMaxDynamicSharedMemorySize, (int)shmem);

  ekpconv_fused<<<NQ_TOT / QB, 256, shmem, stream>>>(
      q_pts, s_pts, neighb, x, q_lrf, s_lrf, weights, kp, mlp_w, mlp_b, out);
}